// SpatialConv2d_75720273429182
// MI455X (gfx1250) — compile-verified
//
#include <hip/hip_runtime.h>

// ---------------------------------------------------------------------------
// SpatialConv2d pipeline for MI455X (gfx1250, wave32).
//   K0: weight transpose into WMMA-fragment-native layout (one-time, tiny)
//   K1: 3x3 conv as implicit GEMM via V_WMMA_F32_16X16X4_F32 (exact fp32),
//       async global->LDS staging, b64 fragment loads, coalesced epilogue
//   K2: per-channel BN statistics (training mode, biased var)
//   K3: codebook distance softmax + BN-normalize + attended + 1x1 conv (sq)
//   K4: 3x3 zero-pad softmax of sq + weighted window statistics
// ---------------------------------------------------------------------------

#define BATCH 8
#define CH_IN 69     // 64 conv channels + 5 shape channels
#define CIN   64
#define COUT  64
#define HH    64
#define WW    64
#define HW    4096
#define NPIX  (BATCH * HW)   // 32768

typedef float v2f __attribute__((ext_vector_type(2)));
typedef float v8f __attribute__((ext_vector_type(8)));

#if __has_builtin(__builtin_amdgcn_global_load_async_to_lds_b32)
#define USE_ASYNC_LDS 1
#else
#define USE_ASYNC_LDS 0
#endif

__device__ __forceinline__ int clampi(int v, int lo, int hi) {
    return v < lo ? lo : (v > hi ? hi : v);
}

// async global->LDS 4B copy (ASYNCcnt-tracked) with synchronous fallback
__device__ __forceinline__ void copy_g2l_f32(const float* g, float* l) {
#if USE_ASYNC_LDS
    __builtin_amdgcn_global_load_async_to_lds_b32(
        (__attribute__((address_space(1))) int*)(void*)g,
        (__attribute__((address_space(3))) int*)(void*)l, 0, 0);
#else
    *l = *g;
#endif
}

__device__ __forceinline__ void wait_async_lds() {
#if USE_ASYNC_LDS
#if __has_builtin(__builtin_amdgcn_s_wait_asynccnt)
    __builtin_amdgcn_s_wait_asynccnt(0);
#else
    asm volatile("s_wait_asynccnt 0" ::: "memory");
#endif
#endif
}

// ---------------------------------------------------------------------------
// K0: transpose conv weights into B-fragment-native layout.
// k = pos*64 + cin (pos-major).  wtp[pair][n][par] with pair = k/2, par = k&1:
//   wtp[pair*128 + n*2 + par] = wgt[n*576 + cin*9 + pos]
// A B-fragment load is then ONE coalesced b64: rows (k, k+1) at column n.
// ---------------------------------------------------------------------------
__global__ __launch_bounds__(256) void weight_prep_kernel(
    const float* __restrict__ wgt, float* __restrict__ wtp)
{
    const int o = blockIdx.x * 256 + threadIdx.x;   // 0 .. 36863
    const int pair = o >> 7;
    const int rem  = o & 127;
    const int n    = rem >> 1;
    const int par  = rem & 1;
    const int k    = 2 * pair + par;
    const int pos  = k >> 6;
    const int cin  = k & 63;
    wtp[o] = wgt[n * 576 + cin * 9 + pos];
}

// ---------------------------------------------------------------------------
// K1: conv3x3 via WMMA. Block = 256 thr (8 waves) -> 2 output rows x 64 couts.
// GEMM: M = pixels (16/wave), N = couts (4 tiles of 16), K = 576 (pos-major).
// LDS tile layout [4 rows][66 cols][32 cins, pad 34] -> A frag = one b64.
// Epilogue transposes C through LDS (stride 133, conflict-free) so the
// conv_raw stores are fully coalesced.
// ---------------------------------------------------------------------------
__global__ __launch_bounds__(256) void conv3x3_wmma_kernel(
    const float* __restrict__ x,      // [B][69][64][64]
    const float* __restrict__ wtp,    // [288 pairs][64 n][2]
    const float* __restrict__ bias,   // [64]
    float* __restrict__ conv_raw)     // [B][64][64][64]
{
    __shared__ float smem[4 * 66 * 34];    // 8976 floats; reused by epilogue

    const int blk  = blockIdx.x;           // b*32 + rowPair
    const int b    = blk >> 5;
    const int oh0  = (blk & 31) * 2;
    const int tid  = threadIdx.x;
    const int wave = tid >> 5;
    const int lane = tid & 31;
    const int lr      = wave >> 2;              // local output row 0/1
    const int colBase = (wave & 3) * 16;        // pixel column base of M-tile
    const int laneN   = lane & 15;
    const int hiK     = (lane >> 4) * 2;        // K offset per ISA A/B layout
    const int hik2    = hiK >> 1;
    const int mHi     = (lane >> 4) << 3;

    // Warm the transposed-weight working set (144 KB -> L2/WGP$).
    __builtin_prefetch(wtp + tid * 144, 0, 0);

    const float* xb = x + (size_t)b * CH_IN * HW;

    v8f acc[4];
    v8f zero = {0.f, 0.f, 0.f, 0.f, 0.f, 0.f, 0.f, 0.f};
#pragma unroll
    for (int nt = 0; nt < 4; ++nt) acc[nt] = zero;

    const int col = colBase + laneN;            // this lane's pixel column (M)

    for (int half = 0; half < 2; ++half) {
        __syncthreads();                        // prior compute done w/ smem
        // Stage 32 cins x 4 rows x 66 cols; global reads coalesced (col-fast),
        // LDS layout [r][col][cin] with pad-34 (conflict-free b64 A frags).
        for (int i = tid; i < 32 * 4 * 66; i += 256) {
            int c   = i / 264;
            int rem = i - c * 264;              // r*66 + col
            int r   = rem / 66;
            int cc  = rem - r * 66;
            int ir  = clampi(oh0 - 1 + r, 0, 63);
            int ic  = clampi(cc - 1, 0, 63);
            copy_g2l_f32(xb + (half * 32 + c) * HW + ir * WW + ic,
                         &smem[rem * 34 + c]);
        }
        wait_async_lds();
        __syncthreads();

        for (int pos = 0; pos < 9; ++pos) {
            const int kh = pos / 3, kw = pos - 3 * (pos / 3);
            const float* arow = &smem[((lr + kh) * 66 + col + kw) * 34 + hiK];
            const float* wrow = wtp + (pos * 32 + half * 16 + hik2) * 128
                              + laneN * 2;
            for (int cg = 0; cg < 8; ++cg) {
                v2f a = *(const v2f*)(arow + cg * 4);       // ds_load_b64
                const float* wp = wrow + cg * 256;          // cg*2 pairs
#pragma unroll
                for (int nt = 0; nt < 4; ++nt) {
                    v2f bf = *(const v2f*)(wp + nt * 32);   // global b64
                    acc[nt] = __builtin_amdgcn_wmma_f32_16x16x4_f32(
                        false, a, false, bf, (short)0, acc[nt], false, false);
                }
            }
        }
    }

    // Epilogue: C layout -> VGPR r: M = r + 8*(lane>=16); N = lane%16.
    // Transpose through LDS (stride 133: bank-conflict-free), then coalesced
    // stores of [64 ch][2 rows x 64 cols] to conv_raw.
    __syncthreads();
#pragma unroll
    for (int nt = 0; nt < 4; ++nt) {
        const int n = nt * 16 + laneN;
        const float bia = bias[n];
        float* lp = &smem[n * 133 + lr * 64 + colBase + mHi];
#pragma unroll
        for (int r = 0; r < 8; ++r) lp[r] = acc[nt][r] + bia;
    }
    __syncthreads();
    float* ob = conv_raw + (size_t)b * COUT * HW + oh0 * WW;
    for (int i = tid; i < 64 * 128; i += 256) {
        int n = i >> 7, pix = i & 127;
        ob[n * HW + pix] = smem[n * 133 + pix];
    }
}

// ---------------------------------------------------------------------------
// K2: BN training statistics. One block per channel.
// ---------------------------------------------------------------------------
__global__ __launch_bounds__(256) void bn_stats_kernel(
    const float* __restrict__ conv_raw, float* __restrict__ mu,
    float* __restrict__ rs)
{
    const int c   = blockIdx.x;
    const int tid = threadIdx.x;
    float s = 0.f, s2 = 0.f;
    for (int b = 0; b < BATCH; ++b) {
        const float* p = conv_raw + ((size_t)(b * COUT + c)) * HW;
        for (int i = tid; i < HW; i += 256) {
            float v = p[i];
            s += v;
            s2 += v * v;
        }
    }
    __shared__ float sh[256], sh2[256];
    sh[tid] = s; sh2[tid] = s2;
    __syncthreads();
    for (int off = 128; off > 0; off >>= 1) {
        if (tid < off) { sh[tid] += sh[tid + off]; sh2[tid] += sh2[tid + off]; }
        __syncthreads();
    }
    if (tid == 0) {
        const float inv = 1.f / (float)(BATCH * HW);
        float m   = sh[0] * inv;
        float var = sh2[0] * inv - m * m;
        mu[c] = m;
        rs[c] = rsqrtf(var + 1e-5f);
    }
}

// ---------------------------------------------------------------------------
// K3: per pixel: shape window s[45] (edge pad, center-subtract ch 0,1),
// softmax over 64 codes of -||s-cb||^2 (3-pass, register-resident),
// attended = relu(bn * w), sq = 1x1 conv of attended.
// ---------------------------------------------------------------------------
__global__ __launch_bounds__(256) void attend_kernel(
    const float* __restrict__ x, const float* __restrict__ cbg,
    const float* __restrict__ conv_raw, const float* __restrict__ mu,
    const float* __restrict__ rs, const float* __restrict__ w2,
    const float* __restrict__ b2, float* __restrict__ out,
    float* __restrict__ sq)
{
    __shared__ float cbl[COUT * 45];
    for (int i = threadIdx.x; i < COUT * 45; i += 256) cbl[i] = cbg[i];
    __syncthreads();

    const int gid = blockIdx.x * 256 + threadIdx.x;   // b*4096 + hw
    const int b   = gid >> 12;
    const int hw  = gid & 4095;
    const int oh  = hw >> 6, ow = hw & 63;

    // gather 5 channels x 3x3 window (edge replication)
    float s[45];
    const float* xb = x + (size_t)b * CH_IN * HW + 64 * HW;
#pragma unroll
    for (int c = 0; c < 5; ++c) {
#pragma unroll
        for (int p = 0; p < 9; ++p) {
            int r  = clampi(oh - 1 + p / 3, 0, 63);
            int cc = clampi(ow - 1 + p % 3, 0, 63);
            s[c * 9 + p] = xb[c * HW + r * WW + cc];
        }
    }
    const float ctr0 = s[4], ctr1 = s[13];
#pragma unroll
    for (int p = 0; p < 9; ++p) { s[p] -= ctr0; s[9 + p] -= ctr1; }

    // pass A: max logit
    float mx = -3.4e38f;
#pragma unroll 1
    for (int j = 0; j < COUT; ++j) {
        float d = 0.f;
        const float* cj = &cbl[j * 45];
#pragma unroll
        for (int k = 0; k < 45; ++k) { float t = s[k] - cj[k]; d += t * t; }
        mx = fmaxf(mx, -d);
    }
    // pass B: sum of exp
    float sum = 0.f;
#pragma unroll 1
    for (int j = 0; j < COUT; ++j) {
        float d = 0.f;
        const float* cj = &cbl[j * 45];
#pragma unroll
        for (int k = 0; k < 45; ++k) { float t = s[k] - cj[k]; d += t * t; }
        sum += __expf(-d - mx);
    }
    const float inv = 1.f / sum;
    // pass C: attended + sq
    float sqv = 0.f;
#pragma unroll 1
    for (int j = 0; j < COUT; ++j) {
        float d = 0.f;
        const float* cj = &cbl[j * 45];
#pragma unroll
        for (int k = 0; k < 45; ++k) { float t = s[k] - cj[k]; d += t * t; }
        const float wj = __expf(-d - mx) * inv;
        const float cn = (conv_raw[((size_t)(b * COUT + j)) * HW + hw] - mu[j]) * rs[j];
        const float att = fmaxf(cn * wj, 0.f);
        out[((size_t)b * CH_IN + j) * HW + hw] = att;
        sqv += att * w2[j];
    }
    sq[b * HW + hw] = sqv + b2[0];
}

// ---------------------------------------------------------------------------
// K4: q = softmax over 3x3 zero-padded window of sq; weighted window stats
// of the 5 shape channels (edge replication) -> output channels 64..68.
// ---------------------------------------------------------------------------
__global__ __launch_bounds__(256) void win_stats_kernel(
    const float* __restrict__ x, const float* __restrict__ sq,
    float* __restrict__ out)
{
    const int gid = blockIdx.x * 256 + threadIdx.x;
    const int b   = gid >> 12;
    const int hw  = gid & 4095;
    const int oh  = hw >> 6, ow = hw & 63;

    const float* sqb = sq + b * HW;
    float qv[9];
#pragma unroll
    for (int p = 0; p < 9; ++p) {
        int r = oh - 1 + p / 3, c = ow - 1 + p % 3;
        qv[p] = (r >= 0 && r < 64 && c >= 0 && c < 64) ? sqb[r * WW + c] : 0.f;
    }
    float mx = qv[0];
#pragma unroll
    for (int p = 1; p < 9; ++p) mx = fmaxf(mx, qv[p]);
    float sum = 0.f;
#pragma unroll
    for (int p = 0; p < 9; ++p) { qv[p] = __expf(qv[p] - mx); sum += qv[p]; }
    const float inv = 1.f / sum;
#pragma unroll
    for (int p = 0; p < 9; ++p) qv[p] *= inv;

    const float* xb = x + (size_t)b * CH_IN * HW;
    float a9[9], b9[9];
    float m0 = 0.f, m1 = 0.f, v1a = 0.f, v1b = 0.f, c1 = 0.f;
#pragma unroll
    for (int p = 0; p < 9; ++p) {
        int r  = clampi(oh - 1 + p / 3, 0, 63);
        int c  = clampi(ow - 1 + p % 3, 0, 63);
        int o  = r * WW + c;
        a9[p] = xb[64 * HW + o];
        b9[p] = xb[65 * HW + o];
        m0  += a9[p] * qv[p];
        m1  += b9[p] * qv[p];
        v1a += xb[66 * HW + o] * qv[p];
        v1b += xb[67 * HW + o] * qv[p];
        c1  += xb[68 * HW + o] * qv[p];
    }
    float v2a = 0.f, v2b = 0.f, c2 = 0.f;
#pragma unroll
    for (int p = 0; p < 9; ++p) {
        float da = a9[p] - m0, db = b9[p] - m1;
        v2a += da * da * qv[p];
        v2b += db * db * qv[p];
        c2  += da * db * qv[p];
    }
    float* ob = out + ((size_t)b * CH_IN + 64) * HW + hw;
    ob[0 * HW] = m0;
    ob[1 * HW] = m1;
    ob[2 * HW] = v1a + v2a;
    ob[3 * HW] = v1b + v2b;
    ob[4 * HW] = c1 + c2;
}

// ---------------------------------------------------------------------------
extern "C" void kernel_launch(void* const* d_in, const int* in_sizes, int n_in,
                              void* d_out, int out_size, void* d_ws, size_t ws_size,
                              hipStream_t stream) {
    (void)in_sizes; (void)n_in; (void)out_size; (void)ws_size;
    const float* x   = (const float*)d_in[0];  // [8][69][64][64]
    const float* cw  = (const float*)d_in[1];  // [64][64][3][3]
    const float* cb  = (const float*)d_in[2];  // [64]
    const float* w2  = (const float*)d_in[3];  // [64] (1x1 conv)
    const float* b2  = (const float*)d_in[4];  // [1]
    const float* shk = (const float*)d_in[5];  // [64][45]
    float* out = (float*)d_out;
    float* ws  = (float*)d_ws;

    float* conv_raw = ws;                       // 2,097,152 floats
    float* mu       = ws + 2097152;             // 64
    float* rs       = mu + 64;                  // 64
    float* sq       = rs + 64;                  // 32768
    float* wtp      = sq + 32768;               // 36864 (transposed weights)

    weight_prep_kernel<<<144, 256, 0, stream>>>(cw, wtp);
    conv3x3_wmma_kernel<<<BATCH * 32, 256, 0, stream>>>(x, wtp, cb, conv_raw);
    bn_stats_kernel<<<COUT, 256, 0, stream>>>(conv_raw, mu, rs);
    attend_kernel<<<NPIX / 256, 256, 0, stream>>>(x, shk, conv_raw, mu, rs,
                                                  w2, b2, out, sq);
    win_stats_kernel<<<NPIX / 256, 256, 0, stream>>>(x, sq, out);
}